// SmoothFilter_90031104459322
// MI455X (gfx1250) — compile-verified
//
#include <hip/hip_runtime.h>
#include <math.h>

// ---------------------------------------------------------------------------
// GCN propagation:  out = D_in^{-1/2} * A * D_out^{-1/2} * X
// N=100000 nodes, E=1.6M edges, d=64 fp32 features.
// Memory-bound, L2-resident (emb table 25.6MB + out 25.6MB << 192MB L2).
// Pipeline (all on `stream`, deterministic, no host sync):
//   1) zero ws counters + output
//   2) u32 atomic degree histograms over src/dst
//   3) per-node 1/sqrt(max(deg,1))
//   4) edge scatter: 16 threads/edge, float4 gather of h[src], 4x
//      global_atomic_add_f32 (inline asm -> guaranteed HW f32 atomic)
//   5) final row-scale as block-diagonal matmul on the matrix core:
//      per 16x16 tile: 4x v_wmma_f32_16x16x4_f32 with A = diag(s) slices.
// ---------------------------------------------------------------------------

typedef __attribute__((ext_vector_type(2))) float v2f;
typedef __attribute__((ext_vector_type(8))) float v8f;

#define DFEAT 64

// ---------------- 1) zero output + degree counters --------------------------
__global__ void zero_kernel(float* __restrict__ out, int n_out,
                            unsigned* __restrict__ cnt, int n_cnt) {
    int i = blockIdx.x * blockDim.x + threadIdx.x;
    if (i < n_out) out[i] = 0.0f;
    if (i < n_cnt) cnt[i] = 0u;
}

// ---------------- 2) degree histograms (u32 HW atomics) ---------------------
__global__ void degree_kernel(const int* __restrict__ src,
                              const int* __restrict__ dst,
                              unsigned* __restrict__ cnt_src,
                              unsigned* __restrict__ cnt_dst, int n_edges) {
    int e = blockIdx.x * blockDim.x + threadIdx.x;
    if (e >= n_edges) return;
    atomicAdd(&cnt_src[src[e]], 1u);   // -> global_atomic_add_u32
    atomicAdd(&cnt_dst[dst[e]], 1u);
}

// ---------------- 3) deg^{-1/2} with clip(deg, 1) ---------------------------
__global__ void inv_sqrt_kernel(const unsigned* __restrict__ cnt_src,
                                const unsigned* __restrict__ cnt_dst,
                                float* __restrict__ s_src,
                                float* __restrict__ s_dst, int n_nodes) {
    int i = blockIdx.x * blockDim.x + threadIdx.x;
    if (i >= n_nodes) return;
    unsigned a = cnt_src[i]; if (a < 1u) a = 1u;
    unsigned b = cnt_dst[i]; if (b < 1u) b = 1u;
    s_src[i] = 1.0f / sqrtf((float)a);   // full-precision to match x**-0.5
    s_dst[i] = 1.0f / sqrtf((float)b);
}

// ---------------- 4) edge gather + f32 atomic scatter -----------------------
// 16 threads per edge; each thread moves one float4 (16B) of the 256B row.
// Inline asm guarantees the CDNA5 non-returning f32 atomic (STOREcnt path),
// never a CAS loop.
__device__ __forceinline__ void gadd_f32(float* p, float v) {
    asm volatile("global_atomic_add_f32 %0, %1, off" :: "v"(p), "v"(v) : "memory");
}
__device__ __forceinline__ void gadd_f32_off4(float* p, float v) {
    asm volatile("global_atomic_add_f32 %0, %1, off offset:4" :: "v"(p), "v"(v) : "memory");
}
__device__ __forceinline__ void gadd_f32_off8(float* p, float v) {
    asm volatile("global_atomic_add_f32 %0, %1, off offset:8" :: "v"(p), "v"(v) : "memory");
}
__device__ __forceinline__ void gadd_f32_off12(float* p, float v) {
    asm volatile("global_atomic_add_f32 %0, %1, off offset:12" :: "v"(p), "v"(v) : "memory");
}

__global__ void edge_scatter_kernel(const float4* __restrict__ emb4,
                                    const int* __restrict__ src,
                                    const int* __restrict__ dst,
                                    const float* __restrict__ s_src,
                                    float* __restrict__ out, int n_edges) {
    int t = blockIdx.x * blockDim.x + threadIdx.x;
    int e = t >> 4;
    if (e >= n_edges) return;
    int l = t & 15;                       // float4 slot within the 64-wide row
    int s = src[e];                       // 16 lanes hit same dword: 1 request
    int d = dst[e];
    float sc = s_src[s];
    float4 v = emb4[(size_t)s * (DFEAT / 4) + l];   // global_load_b128, coalesced
    float* p = out + (size_t)d * DFEAT + l * 4;
    gadd_f32     (p, v.x * sc);
    gadd_f32_off4(p, v.y * sc);
    gadd_f32_off8(p, v.z * sc);
    gadd_f32_off12(p, v.w * sc);
}

// ---------------- 5) row scale via V_WMMA_F32_16X16X4_F32 -------------------
// One wave per 16x16 output tile: D = diag(s_in[n0:n0+16]) * T, split into
// 4 accumulating WMMAs over K=4 slices. fp32 in/out -> bit-compatible with
// a plain multiply. Layouts per ISA 7.12.2:
//   A 16x4 f32 : lane l holds M=l%16, VGPR r holds K = 2*(l/16)+r
//   B 4x16 f32 : VGPR r holds K = r+2*(l/16), N = l%16   (row-striped)
//   C/D 16x16  : VGPR r holds M = r+8*(l/16), N = l%16
// Loads/stores are 64B-contiguous per half-wave -> fully coalesced.
__global__ void scale_wmma_kernel(float* __restrict__ out,
                                  const float* __restrict__ s_dst,
                                  int n_nodes) {
    int gt   = blockIdx.x * blockDim.x + threadIdx.x;
    int wid  = gt >> 5;                 // wave32 -> one tile per wave
    int lane = threadIdx.x & 31;
    int nb = wid >> 2;                  // node block (16 rows)
    int fb = wid & 3;                   // feature block (16 cols of 64)
    int n0 = nb * 16;
    int f0 = fb * 16;
    if (n0 >= n_nodes) return;          // grid sized exactly; no intra-wave div.

    int half = lane >> 4;
    int l16  = lane & 15;
    float sval = s_dst[n0 + l16];

    v8f c = {};
#pragma unroll
    for (int chunk = 0; chunk < 4; ++chunk) {
        int kbase = chunk * 4 + 2 * half;        // K index of VGPR0 this lane
        // B = rows [n0+chunk*4 .. +3] of the tile (the data being scaled)
        v2f b;
        b.x = out[(size_t)(n0 + kbase + 0) * DFEAT + f0 + l16];
        b.y = out[(size_t)(n0 + kbase + 1) * DFEAT + f0 + l16];
        // A = 16x4 slice of diag(s): A[m][k] = s_m iff m == chunk*4+k
        v2f a;
        a.x = (l16 == kbase + 0) ? sval : 0.0f;
        a.y = (l16 == kbase + 1) ? sval : 0.0f;
        c = __builtin_amdgcn_wmma_f32_16x16x4_f32(
                /*neg_a=*/false, a, /*neg_b=*/false, b,
                /*c_mod=*/(short)0, c, /*reuse_a=*/false, /*reuse_b=*/false);
    }
#pragma unroll
    for (int r = 0; r < 8; ++r) {
        out[(size_t)(n0 + r + 8 * half) * DFEAT + f0 + l16] = c[r];
    }
}

// ---------------------------------------------------------------------------
extern "C" void kernel_launch(void* const* d_in, const int* in_sizes, int n_in,
                              void* d_out, int out_size, void* d_ws, size_t ws_size,
                              hipStream_t stream) {
    const float* embeddings = (const float*)d_in[0];
    const int*   src        = (const int*)d_in[1];
    const int*   dst        = (const int*)d_in[2];
    // d_in[3] is n_nodes on-device; derive on host from sizes instead.
    const int n_nodes = in_sizes[0] / DFEAT;
    const int n_edges = in_sizes[1];

    // workspace: cnt_src | cnt_dst | s_src | s_dst  (4 * n_nodes * 4B = 1.6MB)
    unsigned* cnt_src = (unsigned*)d_ws;
    unsigned* cnt_dst = cnt_src + n_nodes;
    float*    s_src   = (float*)(cnt_dst + n_nodes);
    float*    s_dst   = s_src + n_nodes;
    float*    out     = (float*)d_out;

    const int B = 256;

    // 1) zero output + counters
    {
        int n_out = n_nodes * DFEAT;
        int n_cnt = 2 * n_nodes;
        int n = n_out > n_cnt ? n_out : n_cnt;
        zero_kernel<<<(n + B - 1) / B, B, 0, stream>>>(out, n_out, cnt_src, n_cnt);
    }
    // 2) degrees
    degree_kernel<<<(n_edges + B - 1) / B, B, 0, stream>>>(src, dst, cnt_src, cnt_dst, n_edges);
    // 3) deg^{-1/2}
    inv_sqrt_kernel<<<(n_nodes + B - 1) / B, B, 0, stream>>>(cnt_src, cnt_dst, s_src, s_dst, n_nodes);
    // 4) edge gather/scatter (16 threads per edge)
    {
        long long threads = (long long)n_edges * 16;
        edge_scatter_kernel<<<(unsigned)((threads + B - 1) / B), B, 0, stream>>>(
            (const float4*)embeddings, src, dst, s_src, out, n_edges);
    }
    // 5) in-degree scaling on the matrix core (one wave per 16x16 tile)
    {
        int tiles  = (n_nodes / 16) * (DFEAT / 16);   // N divisible by 16 here
        int waves  = tiles;
        int blocks = (waves * 32 + B - 1) / B;
        scale_wmma_kernel<<<blocks, B, 0, stream>>>(out, s_dst, n_nodes);
    }
}